// PairNetLayer_symmetry_exp_84748294684785
// MI455X (gfx1250) — compile-verified
//
#include <hip/hip_runtime.h>
#include <hip/hip_bf16.h>

// ---------------------------------------------------------------------------
// PairNetLayer fused CDNA5 (gfx1250) implementation.
// All GEMMs use V_WMMA_F32_16X16X4_F32 (exact fp32 matrix math, wave32).
// Register-blocked: one A-fragment feeds NB WMMA tiles per k-step.
// ---------------------------------------------------------------------------

#define MULC 128
#define TOTC 1152      // 128 * 9
#define NMC  384       // 3 * 128
#define NTHREADS 128   // 4 wave32s per block

typedef __attribute__((ext_vector_type(2))) float v2f;
typedef __attribute__((ext_vector_type(8))) float v8f;

__constant__ int c_coff[9] = {0, 128, 129, 130, 512, 513, 514, 515, 516};
__constant__ int c_cstr[9] = {1, 3, 3, 3, 5, 5, 5, 5, 5};
__constant__ int c_wsel[9] = {0, 1, 1, 1, 2, 2, 2, 2, 2};

__device__ __forceinline__ v8f wmma_f32_k4(v2f a, v2f b, v8f c) {
  // (neg_a, A, neg_b, B, c_mod, C, reuse_a, reuse_b)
  return __builtin_amdgcn_wmma_f32_16x16x4_f32(
      false, a, false, b, (short)0, c, false, false);
}

// C(16 x 4*ntpw*16) = act(A(16 x K, lds) @ B(K x ldb, global) + bias)
// Wave w owns n-tiles [w*ntpw, (w+1)*ntpw); NB tiles share each A-fragment.
// A frag: lane<16 -> row=lane, K pair {k,k+1}; lane>=16 -> row=lane-16,{k+2,k+3}
// B frag: lane<16 -> rows {k,k+1}, col n0+lane; lane>=16 -> rows {k+2,k+3}
// C/D:    vgpr r -> (M=r, N=lane) lane<16 ; (M=r+8, N=lane-16) lane>=16
template <int NB>
__device__ __forceinline__ void gemm_rowtile_nb(
    const float* __restrict__ As, int lda, int K,
    const float* __restrict__ Bg, int ldb,
    const float* __restrict__ bias,
    float* __restrict__ Cs, int ldc, int ntpw,
    bool do_silu, int wave, int lane) {
  const int arow = lane & 15;
  const int klo  = (lane >> 4) << 1;       // 0 or 2
  const int ncl  = lane & 15;
  const int m0   = (lane >> 4) << 3;       // 0 or 8
  for (int nb = 0; nb < ntpw; nb += NB) {
    const int n0 = (wave * ntpw + nb) << 4;
    const v8f zero = {};
    v8f acc[NB];
#pragma unroll
    for (int j = 0; j < NB; ++j) acc[j] = zero;
    const float* ap = As + arow * lda + klo;       // 8B aligned (lda,klo even)
    const float* bp = Bg + klo * ldb + n0 + ncl;
    for (int k = 0; k < K; k += 4) {
      const v2f a = *(const v2f*)ap;               // ds_load_b64
      ap += 4;
#pragma unroll
      for (int j = 0; j < NB; ++j) {
        v2f b; b.x = bp[j * 16]; b.y = bp[ldb + j * 16];
        acc[j] = wmma_f32_k4(a, b, acc[j]);
      }
      bp += 4 * ldb;
    }
#pragma unroll
    for (int j = 0; j < NB; ++j) {
      const float bi = bias[n0 + j * 16 + ncl];
#pragma unroll
      for (int r = 0; r < 8; ++r) {
        float v = acc[j][r] + bi;
        if (do_silu) v = v / (1.0f + __expf(-v));
        Cs[(m0 + r) * ldc + n0 + j * 16 + ncl] = v;
      }
    }
  }
}

// out(16 x TOT) = irrep_linear(Xs, W)/sqrt(128) + b0(part0) [+ addg]
// Per channel: wave w computes v-tiles {2w, 2w+1}; A-fragment shared by both.
__device__ __forceinline__ void irrep_linear_tile(
    const float* __restrict__ Xs,       // lds, 16 x TOT
    const float* __restrict__ W,        // global, 3 x 128 x 128 (u-major)
    const float* __restrict__ bias0,    // global, 128
    const float* __restrict__ addg,     // nullable global 16 x TOT tile
    float* __restrict__ outg,           // global 16 x TOT tile
    int wave, int lane) {
  const float RS128 = 0.08838834764831845f; // 1/sqrt(128)
  const int arow = lane & 15;
  const int klo  = (lane >> 4) << 1;
  const int ncl  = lane & 15;
  const int m0   = (lane >> 4) << 3;
  const int n0   = wave << 5;             // 32 columns (2 tiles) per wave
  for (int c = 0; c < 9; ++c) {
    const int coff = c_coff[c];
    const int cstr = c_cstr[c];
    const float* Bp = W + c_wsel[c] * (MULC * MULC);
    const v8f zero = {};
    v8f acc0 = zero, acc1 = zero;
    const float* ap = Xs + arow * TOTC + coff + klo * cstr;
    const float* bp = Bp + klo * MULC + n0 + ncl;
    for (int k = 0; k < MULC; k += 4) {
      v2f a;  a.x  = ap[0];  a.y  = ap[cstr];
      v2f b0; b0.x = bp[0];  b0.y = bp[MULC];
      v2f b1; b1.x = bp[16]; b1.y = bp[MULC + 16];
      acc0 = wmma_f32_k4(a, b0, acc0);
      acc1 = wmma_f32_k4(a, b1, acc1);
      ap += 4 * cstr;
      bp += 4 * MULC;
    }
#pragma unroll
    for (int j = 0; j < 2; ++j) {
      const v8f acc = j ? acc1 : acc0;
      const int nn = n0 + j * 16 + ncl;
      const float bi = (c == 0) ? bias0[nn] : 0.0f;
      const int ocol = coff + nn * cstr;
#pragma unroll
      for (int r = 0; r < 8; ++r) {
        float v = acc[r] * RS128 + bi;
        if (addg) v += addg[(m0 + r) * TOTC + ocol];
        outg[(m0 + r) * TOTC + ocol] = v;
      }
    }
  }
}

// f0 = [x0, |part1|, |part2|] per (row, mul)
__device__ __forceinline__ void build_f0(const float* __restrict__ Xs,
                                         float* __restrict__ F0, int tid) {
  for (int i = tid; i < 16 * MULC; i += NTHREADS) {
    const int r = i >> 7, m = i & 127;
    const float* xr = Xs + r * TOTC;
    float s1 = 0.f, s2 = 0.f;
#pragma unroll
    for (int d = 0; d < 3; ++d) { float v = xr[MULC + m * 3 + d]; s1 += v * v; }
#pragma unroll
    for (int d = 0; d < 5; ++d) { float v = xr[4 * MULC + m * 5 + d]; s2 += v * v; }
    float* fr = F0 + r * NMC;
    fr[m]            = xr[m];
    fr[MULC + m]     = sqrtf(s1);
    fr[2 * MULC + m] = sqrtf(s2);
  }
}

// gate in place: part0 <- g0 ; part1 *= g1 ; part2 *= g2
__device__ __forceinline__ void apply_gate(float* __restrict__ Xs,
                                           const float* __restrict__ Gs, int tid) {
  for (int i = tid; i < 16 * MULC; i += NTHREADS) {
    const int r = i >> 7, m = i & 127;
    float* xr = Xs + r * TOTC;
    const float* gr = Gs + r * NMC;
    xr[m] = gr[m];
    const float g1 = gr[MULC + m], g2 = gr[2 * MULC + m];
#pragma unroll
    for (int d = 0; d < 3; ++d) xr[MULC + m * 3 + d] *= g1;
#pragma unroll
    for (int d = 0; d < 5; ++d) xr[4 * MULC + m * 5 + d] *= g2;
  }
}

// ---------------------------------------------------------------------------
// Kernel 1: per 16-node tile -> na0 = irrep(X, W_inner)+b ; na = irrep(gate(X))
// ---------------------------------------------------------------------------
__global__ __launch_bounds__(NTHREADS) void k_nodes(
    const float* __restrict__ node_attr,
    const float* __restrict__ W_inner, const float* __restrict__ b_inner,
    const float* __restrict__ W_n,     const float* __restrict__ b_n,
    const float* __restrict__ pW1, const float* __restrict__ pb1,
    const float* __restrict__ pW2, const float* __restrict__ pb2,
    float* __restrict__ na0, float* __restrict__ na) {
  extern __shared__ float lds[];
  float* Xs = lds;                 // 16*1152
  float* F0 = Xs + 16 * TOTC;      // 16*384  (reused for G)
  float* Hs = F0 + 16 * NMC;       // 16*384
  const int tid = threadIdx.x, lane = tid & 31, wave = tid >> 5;
  const long r0 = (long)blockIdx.x * 16;

  {
    const float4* xg = (const float4*)(node_attr + r0 * TOTC);
    float4* xs4 = (float4*)Xs;
    for (int i = tid; i < 16 * TOTC / 4; i += NTHREADS) xs4[i] = xg[i];
  }
  __syncthreads();

  irrep_linear_tile(Xs, W_inner, b_inner, nullptr, na0 + r0 * TOTC, wave, lane);
  build_f0(Xs, F0, tid);
  __syncthreads();
  gemm_rowtile_nb<6>(F0, NMC, NMC, pW1, NMC, pb1, Hs, NMC, 6, true,  wave, lane);
  __syncthreads();
  gemm_rowtile_nb<6>(Hs, NMC, NMC, pW2, NMC, pb2, F0, NMC, 6, false, wave, lane);
  __syncthreads();
  apply_gate(Xs, F0, tid);
  __syncthreads();
  irrep_linear_tile(Xs, W_n, b_n, nullptr, na + r0 * TOTC, wave, lane);
}

// ---------------------------------------------------------------------------
// Kernel 2: per 16-edge tile -> w = we * ws  (E x 128)
// ---------------------------------------------------------------------------
__global__ __launch_bounds__(NTHREADS) void k_edges_w(
    const float* __restrict__ na0,
    const int* __restrict__ dst, const int* __restrict__ src,
    const float* __restrict__ edge_attr,
    const float* __restrict__ fcE_W1, const float* __restrict__ fcE_W2,
    const float* __restrict__ fc_W1, const float* __restrict__ fc_b1,
    const float* __restrict__ fc_W2, const float* __restrict__ fc_b2,
    float* __restrict__ wbuf) {
  const float RS3 = 0.5773502691896258f;   // 1/sqrt(3)
  const float RS5 = 0.4472135954999579f;   // 1/sqrt(5)
  const float RSH = 0.35355339059327373f;  // 1/sqrt(8)
  extern __shared__ float lds[];
  float* S0  = lds;                 // 16*384
  float* H1  = S0 + 16 * NMC;       // 16*128
  float* WS  = H1 + 16 * MULC;      // 16*128
  float* HID = WS + 16 * MULC;      // 16*8
  __shared__ int sd[16], ss[16];
  const int tid = threadIdx.x, lane = tid & 31, wave = tid >> 5;
  const long e0 = (long)blockIdx.x * 16;

  if (tid < 16) { sd[tid] = dst[e0 + tid]; ss[tid] = src[e0 + tid]; }
  __syncthreads();

  for (int i = tid; i < 16 * NMC; i += NTHREADS) {
    const int r = i / NMC, cc = i - r * NMC;
    const float* pd = na0 + (long)sd[r] * TOTC;
    const float* ps = na0 + (long)ss[r] * TOTC;
    float v;
    if (cc < MULC) {
      v = 0.5f * (pd[cc] + ps[cc]);
    } else if (cc < 2 * MULC) {
      const int m = cc - MULC; float s = 0.f;
#pragma unroll
      for (int d = 0; d < 3; ++d) s += pd[MULC + m * 3 + d] * ps[MULC + m * 3 + d];
      v = s * RS3;
    } else {
      const int m = cc - 2 * MULC; float s = 0.f;
#pragma unroll
      for (int d = 0; d < 5; ++d) s += pd[4 * MULC + m * 5 + d] * ps[4 * MULC + m * 5 + d];
      v = s * RS5;
    }
    S0[i] = v;
  }
  for (int i = tid; i < 16 * 8; i += NTHREADS) {
    const int r = i >> 3, j = i & 7;
    const float* ea = edge_attr + (e0 + r) * 64;
    float s = 0.f;
    for (int k = 0; k < 64; ++k) s += ea[k] * fcE_W1[k * 8 + j];
    s *= 0.125f;                                           // 1/sqrt(64)
    const float sp = (s > 20.f) ? s : __logf(1.f + __expf(s));
    HID[i] = sp - 0.6931471805599453f;                     // ssp
  }
  __syncthreads();
  gemm_rowtile_nb<2>(S0, NMC, NMC,  fc_W1, MULC, fc_b1, H1, MULC, 2, true,  wave, lane);
  __syncthreads();
  gemm_rowtile_nb<2>(H1, MULC, MULC, fc_W2, MULC, fc_b2, WS, MULC, 2, false, wave, lane);
  __syncthreads();
  for (int i = tid; i < 16 * MULC; i += NTHREADS) {
    const int r = i >> 7, n = i & 127;
    float we = 0.f;
#pragma unroll
    for (int j = 0; j < 8; ++j) we += HID[r * 8 + j] * fcE_W2[j * MULC + n];
    wbuf[e0 * MULC + i] = we * RSH * WS[i];
  }
}

// ---------------------------------------------------------------------------
// Kernel 3: per 16-edge tile -> out = irrep(gate((na[s]+na[d])*w_full)) + res
// ---------------------------------------------------------------------------
__global__ __launch_bounds__(NTHREADS) void k_edges_out(
    const float* __restrict__ na,
    const int* __restrict__ dst, const int* __restrict__ src,
    const float* __restrict__ wbuf, const float* __restrict__ node_pair_attr,
    const float* __restrict__ gW1, const float* __restrict__ gb1,
    const float* __restrict__ gW2, const float* __restrict__ gb2,
    const float* __restrict__ W_out, const float* __restrict__ b_out,
    float* __restrict__ out) {
  extern __shared__ float lds[];
  float* Xs = lds;                 // 16*1152 node_pair tile (gated in place)
  float* F0 = Xs + 16 * TOTC;      // 16*384  (reused for G)
  float* Hs = F0 + 16 * NMC;       // 16*384
  float* Wt = Hs + 16 * NMC;       // 16*128  staged w rows
  __shared__ int sd[16], ss[16];
  const int tid = threadIdx.x, lane = tid & 31, wave = tid >> 5;
  const long e0 = (long)blockIdx.x * 16;

  if (tid < 16) { sd[tid] = dst[e0 + tid]; ss[tid] = src[e0 + tid]; }
  {
    const float4* wg = (const float4*)(wbuf + e0 * MULC);
    float4* wt4 = (float4*)Wt;
    for (int i = tid; i < 16 * MULC / 4; i += NTHREADS) wt4[i] = wg[i];
  }
  __syncthreads();

  // node_pair[e, c] = (na[src,c] + na[dst,c]) * w[e, c/9]   (b128 gathers)
  {
    const float4* na4 = (const float4*)na;
    float4* xs4 = (float4*)Xs;
    const int TQ = TOTC / 4;
    for (int i = tid; i < 16 * TQ; i += NTHREADS) {
      const int r = i / TQ, c4 = i - r * TQ;
      const int cc = c4 * 4;
      const float4 a = na4[(long)ss[r] * TQ + c4];
      const float4 b = na4[(long)sd[r] * TQ + c4];
      const float* wr = Wt + r * MULC;
      float4 o;
      o.x = (a.x + b.x) * wr[(cc + 0) / 9];
      o.y = (a.y + b.y) * wr[(cc + 1) / 9];
      o.z = (a.z + b.z) * wr[(cc + 2) / 9];
      o.w = (a.w + b.w) * wr[(cc + 3) / 9];
      xs4[i] = o;
    }
  }
  __syncthreads();
  build_f0(Xs, F0, tid);
  __syncthreads();
  gemm_rowtile_nb<6>(F0, NMC, NMC, gW1, NMC, gb1, Hs, NMC, 6, true,  wave, lane);
  __syncthreads();
  gemm_rowtile_nb<6>(Hs, NMC, NMC, gW2, NMC, gb2, F0, NMC, 6, false, wave, lane);
  __syncthreads();
  apply_gate(Xs, F0, tid);
  __syncthreads();
  irrep_linear_tile(Xs, W_out, b_out, node_pair_attr + e0 * TOTC,
                    out + e0 * TOTC, wave, lane);
}

// ---------------------------------------------------------------------------
extern "C" void kernel_launch(void* const* d_in, const int* in_sizes, int n_in,
                              void* d_out, int out_size, void* d_ws, size_t ws_size,
                              hipStream_t stream) {
  const float* node_attr      = (const float*)d_in[0];
  const float* edge_attr      = (const float*)d_in[1];
  const int*   dst            = (const int*)d_in[2];
  const int*   src            = (const int*)d_in[3];
  const float* node_pair_attr = (const float*)d_in[4];
  const float* W_inner = (const float*)d_in[5];
  const float* b_inner = (const float*)d_in[6];
  const float* W_n     = (const float*)d_in[7];
  const float* b_n     = (const float*)d_in[8];
  const float* W_out   = (const float*)d_in[9];
  const float* b_out   = (const float*)d_in[10];
  const float* pW1 = (const float*)d_in[11];
  const float* pb1 = (const float*)d_in[12];
  const float* pW2 = (const float*)d_in[13];
  const float* pb2 = (const float*)d_in[14];
  const float* gW1 = (const float*)d_in[15];
  const float* gb1 = (const float*)d_in[16];
  const float* gW2 = (const float*)d_in[17];
  const float* gb2 = (const float*)d_in[18];
  const float* fcE_W1 = (const float*)d_in[19];
  const float* fcE_W2 = (const float*)d_in[20];
  const float* fc_W1  = (const float*)d_in[21];
  const float* fc_b1  = (const float*)d_in[22];
  const float* fc_W2  = (const float*)d_in[23];
  const float* fc_b2  = (const float*)d_in[24];

  const int N = in_sizes[0] / TOTC;   // 4096
  const int E = in_sizes[2];          // 65536

  float* na0  = (float*)d_ws;
  float* na   = na0 + (size_t)N * TOTC;
  float* wbuf = na  + (size_t)N * TOTC;

  const dim3 blk(NTHREADS);
  const size_t sh_node = (size_t)(16 * TOTC + 2 * 16 * NMC) * sizeof(float);
  const size_t sh_w    = (size_t)(16 * NMC + 2 * 16 * MULC + 16 * 8) * sizeof(float);
  const size_t sh_out  = (size_t)(16 * TOTC + 2 * 16 * NMC + 16 * MULC) * sizeof(float);

  k_nodes<<<N / 16, blk, sh_node, stream>>>(node_attr, W_inner, b_inner, W_n, b_n,
                                            pW1, pb1, pW2, pb2, na0, na);
  k_edges_w<<<E / 16, blk, sh_w, stream>>>(na0, dst, src, edge_attr, fcE_W1, fcE_W2,
                                           fc_W1, fc_b1, fc_W2, fc_b2, wbuf);
  k_edges_out<<<E / 16, blk, sh_out, stream>>>(na, dst, src, wbuf, node_pair_attr,
                                               gW1, gb1, gW2, gb2, W_out, b_out,
                                               (float*)d_out);
}